// RoIPool_24060406792407
// MI455X (gfx1250) — compile-verified
//
#include <hip/hip_runtime.h>
#include <float.h>
#include <stdint.h>

// ---------------- problem constants (from reference setup_inputs) ------------
#define P_   7        // pooled 7x7
#define CH_  8        // channels per block (== waves per block)
#define C_   512
#define H_   38
#define W_   50
#define SCALE_ 0.0625f
#define LDS_STRIDE_ 1904   // per-channel LDS pitch in floats (16B-aligned, >= 38*50+2)

// ---------------- CDNA5 async global->LDS path, guarded ----------------------
#if defined(__gfx1250__) && \
    __has_builtin(__builtin_amdgcn_global_load_async_to_lds_b32) && \
    __has_builtin(__builtin_amdgcn_global_load_async_to_lds_b128) && \
    __has_builtin(__builtin_amdgcn_s_wait_asynccnt)
#define USE_ASYNC_LDS 1
#else
#define USE_ASYNC_LDS 0
#endif

// clang-22 async-LDS builtin param types:
//   b32 : (AS1 int*, AS3 int*, imm, imm)
//   b128: (AS1 v4i*, AS3 v4i*, imm, imm)   v4i = int __attribute__((vector_size(16)))
typedef int v4i_ __attribute__((vector_size(16)));
typedef __attribute__((address_space(1))) int  GINT;   // global scalar (AS1)
typedef __attribute__((address_space(3))) int  LINT;   // LDS scalar (AS3)
typedef __attribute__((address_space(1))) v4i_ GV4;    // global v4i (AS1)
typedef __attribute__((address_space(3))) v4i_ LV4;    // LDS v4i (AS3)

__global__ __launch_bounds__(256) void roipool_cdna5_kernel(
    const float* __restrict__ feat,   // [N,C,H,W]
    const float* __restrict__ rois,   // [R,5] = (b,x1,y1,x2,y2)
    float* __restrict__ out)          // [R,C,7,7]
{
    __shared__ float smem[CH_ * LDS_STRIDE_];      // 8 * 1904 * 4B = 60928 B

    const int r  = blockIdx.y;                     // roi index
    const int c0 = blockIdx.x * CH_;               // first channel of this block

    // ---- RoI decode (uniform across the block; RNE rounding == jnp.round) ----
    const float* roi = rois + (size_t)r * 5;
    const int b   = (int)roi[0];
    const int rsw = (int)rintf(roi[1] * SCALE_);
    const int rsh = (int)rintf(roi[2] * SCALE_);
    const int rew = (int)rintf(roi[3] * SCALE_);
    const int reh = (int)rintf(roi[4] * SCALE_);
    const int roi_h = max(reh - rsh + 1, 1);
    const int roi_w = max(rew - rsw + 1, 1);
    const float bin_h = (float)roi_h / (float)P_;
    const float bin_w = (float)roi_w / (float)P_;

    // union of all bin row-windows: [h0, h1)
    const int h0 = min(max(rsh, 0), H_);
    const int h1 = min(max(rsh + roi_h, 0), H_);
    const int wh = h1 - h0;                        // staged row count (0..38)

    const int wave = threadIdx.x >> 5;
    const int lane = threadIdx.x & 31;

    // ---- stage: one wave per channel. Rows [h0,h1) of one channel plane are a
    //      single contiguous global block of wh*50 floats. Align the copy start
    //      down to 16B (pad = 0 or 2 floats since h0*50 is always even) and use
    //      b128 asyncs for the bulk, predicated b32 asyncs for the <=2-float
    //      tail. b128 count is floored so we never read past the plane. ----
    const int pad = (h0 * W_) & 3;                 // 0 or 2 floats
    {
        const int c = c0 + wave;
        const float* g0 = feat + (size_t)(b * C_ + c) * (H_ * W_)
                               + (size_t)(h0 * W_ - pad);      // 16B-aligned
        float* l0 = smem + wave * LDS_STRIDE_;                 // 16B-aligned
        const int ntot = wh * W_ + pad;            // floats to copy
#if USE_ASYNC_LDS
        const int nb4 = ntot >> 2;                 // full 16B chunks
        const int rem = ntot & 3;                  // 0 or 2 tail floats
        for (int i = lane; i < nb4; i += 32) {
            __builtin_amdgcn_global_load_async_to_lds_b128(
                (GV4*)(uintptr_t)(g0 + 4 * i),
                (LV4*)(uint32_t)(uintptr_t)(l0 + 4 * i),
                /*imm offset*/0, /*cpol*/0);
        }
        if (lane < rem) {
            __builtin_amdgcn_global_load_async_to_lds_b32(
                (GINT*)(uintptr_t)(g0 + 4 * nb4 + lane),
                (LINT*)(uint32_t)(uintptr_t)(l0 + 4 * nb4 + lane),
                /*imm offset*/0, /*cpol*/0);
        }
        __builtin_amdgcn_s_wait_asynccnt(0);
#else
        for (int i = lane; i < ntot; i += 32) l0[i] = g0[i];
#endif
    }
    __syncthreads();

    // ---- compute: 8 channels x 49 bins = 392 outputs per block ----
    for (int o = threadIdx.x; o < CH_ * P_ * P_; o += 256) {
        const int c   = o / (P_ * P_);
        const int bin = o - c * (P_ * P_);
        const int ph  = bin / P_;
        const int pw  = bin - ph * P_;

        int hs = min(max((int)floorf((float)ph * bin_h)        + rsh, 0), H_);
        int he = min(max((int)ceilf ((float)(ph + 1) * bin_h)  + rsh, 0), H_);
        int ws = min(max((int)floorf((float)pw * bin_w)        + rsw, 0), W_);
        int we = min(max((int)ceilf ((float)(pw + 1) * bin_w)  + rsw, 0), W_);

        float m;
        if (he <= hs || we <= ws) {
            m = 0.0f;                               // empty bin -> 0 (reference)
        } else {
            m = -FLT_MAX;
            // element (h,w) lives at smem[c*STRIDE + pad + (h - h0)*50 + w]
            const float* base = smem + c * LDS_STRIDE_ + pad - h0 * W_;
            for (int h = hs; h < he; ++h) {
                const float* row = base + h * W_;
                for (int w = ws; w < we; ++w) m = fmaxf(m, row[w]);
            }
        }
        // out[r][c0+c][ph][pw]: lane-contiguous -> coalesced; non-temporal so
        // the 200MB output stream does not evict the L2-resident features.
        __builtin_nontemporal_store(
            m, &out[((size_t)r * C_ + (c0 + c)) * (P_ * P_) + bin]);
    }
}

extern "C" void kernel_launch(void* const* d_in, const int* in_sizes, int n_in,
                              void* d_out, int out_size, void* d_ws, size_t ws_size,
                              hipStream_t stream) {
    const float* feat = (const float*)d_in[0];   // [4,512,38,50] f32
    const float* rois = (const float*)d_in[1];   // [R,5] f32
    float* out = (float*)d_out;                  // [R,512,7,7] f32
    const int R = in_sizes[1] / 5;               // 2000

    dim3 grid(C_ / CH_, R);                      // (64, 2000)
    roipool_cdna5_kernel<<<grid, 256, 0, stream>>>(feat, rois, out);
    (void)n_in; (void)out_size; (void)d_ws; (void)ws_size;
}